// SAGELayer_30442728194383
// MI455X (gfx1250) — compile-verified
//
#include <hip/hip_runtime.h>

typedef __attribute__((ext_vector_type(2))) float v2f;
typedef __attribute__((ext_vector_type(8))) float v8f;

#define NN 50000      // nodes
#define NE 800000     // edges
#define DIN 128       // node feat dim (== edge dim == out dim)
#define LSTR 260      // LDS row stride in floats (4-float skew -> conflict-free b64 reads)

// ---------------- zero workspace (sum + cnt) ----------------
__global__ void zero_ws_k(float* __restrict__ p, int n) {
    int i = blockIdx.x * blockDim.x + threadIdx.x;
    if (i < n) p[i] = 0.0f;
}

// ---------------- segment sum of edge feats onto dst ----------------
// one thread per (edge, 4 feats): float4 load + 4 f32 atomics
__global__ void scatter_sum_k(const float* __restrict__ efeats,
                              const int* __restrict__ dst,
                              float* __restrict__ sum,
                              float* __restrict__ cnt) {
    long i = (long)blockIdx.x * blockDim.x + threadIdx.x;   // i in [0, NE*32)
    if (i >= (long)NE * 32) return;
    int e  = (int)(i >> 5);
    int f4 = (int)(i & 31);
    int d  = dst[e];
    float4 v = *(const float4*)(efeats + (long)e * DIN + f4 * 4);
    float* s = sum + (long)d * DIN + f4 * 4;
    atomicAdd(s + 0, v.x);
    atomicAdd(s + 1, v.y);
    atomicAdd(s + 2, v.z);
    atomicAdd(s + 3, v.w);
    if (f4 == 0) atomicAdd(&cnt[d], 1.0f);
}

// ---------------- node update: h = relu([nfeats, sum/cnt] @ Wa^T + ba) ----
// block = 256 threads = 8 waves -> one 16-node tile; wave w -> 16-col tile w.
// Stage concat rows (mean folded in) into LDS, then branch-free WMMA loop.
__global__ void node_update_k(const float* __restrict__ nfeats,
                              const float* __restrict__ sum,
                              const float* __restrict__ cnt,
                              const float* __restrict__ Wa,   // [128, 256]
                              const float* __restrict__ ba,   // [128]
                              float* __restrict__ h) {        // [NN, 128]
    __shared__ float lds[16 * LSTR];
    const int wave = threadIdx.x >> 5;
    const int lane = threadIdx.x & 31;
    const int tile = blockIdx.x;

    // cooperative stage: wave w stages rows w and w+8
#pragma unroll
    for (int rr = 0; rr < 2; ++rr) {
        const int r    = wave + rr * 8;
        const int node = tile * 16 + r;
        const float inv = 1.0f / fmaxf(cnt[node], 1.0f);
        float4 v0 = *(const float4*)(nfeats + (long)node * DIN + lane * 4);
        float4 v1 = *(const float4*)(sum    + (long)node * DIN + lane * 4);
        v1.x *= inv; v1.y *= inv; v1.z *= inv; v1.w *= inv;
        *(float4*)&lds[r * LSTR + lane * 4]       = v0;
        *(float4*)&lds[r * LSTR + 128 + lane * 4] = v1;
    }
    __syncthreads();

    const int row = lane & 15;             // A row (M) / B col (N) for this lane
    const int kk  = (lane >> 4) << 1;      // K sub-offset: 0 or 2
    const float* arow = &lds[row * LSTR];
    const float* brow = Wa + (long)(wave * 16 + row) * 256;

    v8f acc = {};
#pragma unroll 8
    for (int k0 = 0; k0 < 256; k0 += 4) {
        v2f a = *(const v2f*)(arow + k0 + kk);   // ds_load_b64
        v2f b = *(const v2f*)(brow + k0 + kk);   // global_load_b64 (cache-resident)
        acc = __builtin_amdgcn_wmma_f32_16x16x4_f32(
                  false, a, false, b, (short)0, acc, false, false);
    }
    const int  ncol  = wave * 16 + row;
    const long mbase = (long)tile * 16 + ((lane >> 4) << 3);
    const float bv   = ba[ncol];
#pragma unroll
    for (int r = 0; r < 8; ++r) {
        float v = acc[r] + bv;
        h[(mbase + r) * DIN + ncol] = fmaxf(v, 0.0f);
    }
}

// ---------------- edge MLP: edge = [h[src], h[dst]] @ We^T + be -----------
// block = 512 threads = 16 waves -> one 16-edge tile; wave w -> 16-col tile w.
__global__ void edge_mlp_k(const float* __restrict__ h,
                           const int* __restrict__ src,
                           const int* __restrict__ dst,
                           const float* __restrict__ We,   // [256, 256]
                           const float* __restrict__ be,   // [256]
                           float* __restrict__ out) {      // [NE, 256]
    __shared__ float lds[16 * LSTR];
    const int wave = threadIdx.x >> 5;
    const int lane = threadIdx.x & 31;
    const int tile = blockIdx.x;

    // cooperative stage: wave w gathers edge (tile*16+w)'s concat row
    {
        const int e = tile * 16 + wave;
        const int s = src[e];
        const int d = dst[e];
        float4 v0 = *(const float4*)(h + (long)s * DIN + lane * 4);
        float4 v1 = *(const float4*)(h + (long)d * DIN + lane * 4);
        *(float4*)&lds[wave * LSTR + lane * 4]       = v0;
        *(float4*)&lds[wave * LSTR + 128 + lane * 4] = v1;
    }
    __syncthreads();

    const int row = lane & 15;
    const int kk  = (lane >> 4) << 1;
    const float* arow = &lds[row * LSTR];
    const float* brow = We + (long)(wave * 16 + row) * 256;

    v8f acc = {};
#pragma unroll 8
    for (int k0 = 0; k0 < 256; k0 += 4) {
        v2f a = *(const v2f*)(arow + k0 + kk);   // ds_load_b64, conflict-free
        v2f b = *(const v2f*)(brow + k0 + kk);   // We streams via WGP$/L2
        acc = __builtin_amdgcn_wmma_f32_16x16x4_f32(
                  false, a, false, b, (short)0, acc, false, false);
    }
    const int  ncol  = wave * 16 + row;
    const long mbase = (long)tile * 16 + ((lane >> 4) << 3);
    const float bv   = be[ncol];
#pragma unroll
    for (int r = 0; r < 8; ++r) {
        out[(mbase + r) * 256 + ncol] = acc[r] + bv;   // coalesced 64B/halfwave
    }
}

extern "C" void kernel_launch(void* const* d_in, const int* in_sizes, int n_in,
                              void* d_out, int out_size, void* d_ws, size_t ws_size,
                              hipStream_t stream) {
    const float* nfeats = (const float*)d_in[0];
    const float* efeats = (const float*)d_in[1];
    const int*   src    = (const int*)d_in[2];
    const int*   dst    = (const int*)d_in[3];
    const float* Wa     = (const float*)d_in[4];
    const float* ba     = (const float*)d_in[5];
    const float* We     = (const float*)d_in[6];
    const float* be     = (const float*)d_in[7];

    float* h    = (float*)d_out;                       // [NN,128]
    float* edge = (float*)d_out + (long)NN * DIN;      // [NE,256]

    float* sum = (float*)d_ws;                         // [NN,128]
    float* cnt = sum + (long)NN * DIN;                 // [NN]

    // 1) zero accumulators (ws is poisoned by harness)
    const int nz = NN * DIN + NN;
    zero_ws_k<<<(nz + 255) / 256, 256, 0, stream>>>(sum, nz);

    // 2) segment sum + degree count (float4 per thread)
    const long na = (long)NE * 32;
    scatter_sum_k<<<(int)((na + 255) / 256), 256, 0, stream>>>(efeats, dst, sum, cnt);

    // 3) node GEMM + relu  (50000/16 = 3125 tiles exactly)
    node_update_k<<<NN / 16, 256, 0, stream>>>(nfeats, sum, cnt, Wa, ba, h);

    // 4) edge GEMM (800000/16 = 50000 tiles exactly)
    edge_mlp_k<<<NE / 16, 512, 0, stream>>>(h, src, dst, We, be, edge);
}